// Struct_Feat_Pretrain_5944234737812
// MI455X (gfx1250) — compile-verified
//
#include <hip/hip_runtime.h>

#define N_NODES 100000
#define N_EDGES 1600000
#define DIM     128
#define NLAYERS 3

typedef __attribute__((ext_vector_type(2))) float v2f;
typedef __attribute__((ext_vector_type(8))) float v8f;

// ---------------------------------------------------------------- utilities
__global__ void k_zero_f4(float4* __restrict__ p, int n4) {
    int i = blockIdx.x * blockDim.x + threadIdx.x;
    if (i < n4) p[i] = make_float4(0.f, 0.f, 0.f, 0.f);
}

__global__ void k_copy_f4(const float4* __restrict__ in, float4* __restrict__ out, int n4) {
    int i = blockIdx.x * blockDim.x + threadIdx.x;
    if (i < n4) out[i] = in[i];
}

// ---------------------------------------------------------------- edge scatter
// One wave per edge: 32 lanes x float4 = 128 dims. Gather h[src] (coalesced
// 512B row, L2-resident) and atomic-add into agg[dst] (L2-resident).
__global__ void k_scatter(const float* __restrict__ h,
                          const int* __restrict__ src,
                          const int* __restrict__ dst,
                          float* __restrict__ agg) {
    int gid  = blockIdx.x * blockDim.x + threadIdx.x;
    int edge = gid >> 5;
    int lane = gid & 31;
    if (edge >= N_EDGES) return;
    int s = src[edge];
    int d = dst[edge];
    const float4 v = ((const float4*)(h + (size_t)s * DIM))[lane];
    float* ap = agg + (size_t)d * DIM + (size_t)lane * 4;
    atomicAdd(ap + 0, v.x);
    atomicAdd(ap + 1, v.y);
    atomicAdd(ap + 2, v.z);
    atomicAdd(ap + 3, v.w);
}

// ---------------------------------------------------------------- WMMA GEMM
// y[m0..m0+15][:] = ((1+eps)*h + agg)[rows] @ W[layer] + b[layer]
// Block = 256 threads = 8 waves; wave w owns output columns [16w, 16w+16).
// K-loop: 32 x V_WMMA_F32_16X16X4_F32 (full fp32 accumulate).
__global__ void k_gin_gemm_wmma(const float* __restrict__ h,
                                const float* __restrict__ agg,
                                const float* __restrict__ Wmat,
                                const float* __restrict__ bias,
                                const float* __restrict__ epsArr,
                                int layer,
                                float* __restrict__ y) {
    __shared__ float xt[16 * 132];   // stride 132 avoids LDS bank conflicts

    const int tid = threadIdx.x;
    const int m0  = blockIdx.x * 16;
    const float eps1 = 1.0f + epsArr[layer];

    // Stage x tile: x = (1+eps)*h + agg for 16 nodes x 128 dims
    #pragma unroll
    for (int i = 0; i < 8; ++i) {
        int idx = tid + i * 256;
        int r = idx >> 7;
        int c = idx & 127;
        size_t g = (size_t)(m0 + r) * DIM + c;
        xt[r * 132 + c] = eps1 * h[g] + agg[g];
    }
    __syncthreads();

    const int wave = tid >> 5;
    const int lane = tid & 31;
    const int n0   = wave * 16;
    const int ln   = lane & 15;          // M index for A-frag / N index for B-frag
    const int koff = (lane >> 4) << 1;   // lanes 16-31 hold K+2,K+3

    const float* Wl = Wmat + (size_t)layer * DIM * DIM;

    v8f acc = {};
    for (int k0 = 0; k0 < DIM; k0 += 4) {
        v2f a, b;
        // A 16x4 fp32 layout: VGPR0 = K(koff), VGPR1 = K(koff+1), lane%16 = M
        a.x = xt[ln * 132 + k0 + koff];
        a.y = xt[ln * 132 + k0 + koff + 1];
        // B 4x16 fp32 layout (mirror): lane%16 = N, half-wave selects K pair
        b.x = Wl[(size_t)(k0 + koff)     * DIM + n0 + ln];
        b.y = Wl[(size_t)(k0 + koff + 1) * DIM + n0 + ln];
        acc = __builtin_amdgcn_wmma_f32_16x16x4_f32(
            /*neg_a=*/false, a, /*neg_b=*/false, b,
            /*c_mod=*/(short)0, acc, /*reuse_a=*/false, /*reuse_b=*/false);
    }

    // C/D layout: VGPR r -> row (r or r+8), lane%16 -> column
    const int mrow = m0 + (lane >> 4) * 8;
    const int col  = n0 + ln;
    const float bc = bias[(size_t)layer * DIM + col];
    #pragma unroll
    for (int r = 0; r < 8; ++r) {
        y[(size_t)(mrow + r) * DIM + col] = acc[r] + bc;
    }
}

// ---------------------------------------------------------------- BN column stats
// 128 threads/block (thread = column, coalesced reads), 200 blocks x 500 rows.
__global__ void k_col_stats(const float* __restrict__ x,
                            float* __restrict__ sum,
                            float* __restrict__ sumsq) {
    const int col = threadIdx.x;
    const int r0  = blockIdx.x * 500;
    float s = 0.f, ss = 0.f;
    for (int r = r0; r < r0 + 500; ++r) {
        float v = x[(size_t)r * DIM + col];
        s  += v;
        ss += v * v;
    }
    atomicAdd(&sum[col],   s);
    atomicAdd(&sumsq[col], ss);
}

// ---------------------------------------------------------------- BN + ReLU (in place)
__global__ void k_bn_relu4(float4* __restrict__ x,
                           const float* __restrict__ sum,
                           const float* __restrict__ sumsq,
                           const float* __restrict__ gamma,
                           const float* __restrict__ beta,
                           int layer) {
    const float invN = 1.0f / (float)N_NODES;
    int i = blockIdx.x * blockDim.x + threadIdx.x;      // over N*D/4
    if (i >= N_NODES * DIM / 4) return;
    int c0 = (i * 4) & 127;

    float4 v = x[i];
    float* vp = &v.x;
    #pragma unroll
    for (int j = 0; j < 4; ++j) {
        int c = c0 + j;
        float mean = sum[c] * invN;
        float var  = sumsq[c] * invN - mean * mean;      // biased variance
        float g  = gamma[(size_t)layer * DIM + c];
        float bt = beta [(size_t)layer * DIM + c];
        float z  = (vp[j] - mean) * rsqrtf(var + 1e-5f) * g + bt;
        vp[j] = z > 0.f ? z : 0.f;
    }
    x[i] = v;
}

// ---------------------------------------------------------------- launch
extern "C" void kernel_launch(void* const* d_in, const int* in_sizes, int n_in,
                              void* d_out, int out_size, void* d_ws, size_t ws_size,
                              hipStream_t stream) {
    const float* features = (const float*)d_in[0];
    const float* W        = (const float*)d_in[1];
    const float* b        = (const float*)d_in[2];
    const float* eps      = (const float*)d_in[3];
    const float* g1       = (const float*)d_in[4];
    const float* be1      = (const float*)d_in[5];
    const float* g2       = (const float*)d_in[6];
    const float* be2      = (const float*)d_in[7];
    const int*   src      = (const int*)d_in[8];
    const int*   dst      = (const int*)d_in[9];
    float* out = (float*)d_out;

    float* agg   = (float*)d_ws;                      // N*D floats (51.2 MB)
    float* stats = agg + (size_t)N_NODES * DIM;       // 512 floats: s1,ss1,s2,ss2

    const int ND4   = N_NODES * DIM / 4;              // 3,200,000
    const int CPBLK = 256;

    // h0 = features -> output slot 0
    k_copy_f4<<<(ND4 + CPBLK - 1) / CPBLK, CPBLK, 0, stream>>>(
        (const float4*)features, (float4*)out, ND4);

    for (int L = 0; L < NLAYERS; ++L) {
        const float* hcur  = out + (size_t)L       * N_NODES * DIM;
        float*       hnext = out + (size_t)(L + 1) * N_NODES * DIM;

        // zero agg + stat accumulators
        k_zero_f4<<<(ND4 + CPBLK - 1) / CPBLK, CPBLK, 0, stream>>>((float4*)agg, ND4);
        k_zero_f4<<<1, 128, 0, stream>>>((float4*)stats, 512 / 4);

        // agg[dst] += h[src]   (1 wave per edge)
        k_scatter<<<(N_EDGES * 32) / CPBLK, CPBLK, 0, stream>>>(hcur, src, dst, agg);

        // y = ((1+eps)h + agg) @ W + b   -> write into output slot L+1
        k_gin_gemm_wmma<<<N_NODES / 16, 256, 0, stream>>>(hcur, agg, W, b, eps, L, hnext);

        // BN1 + ReLU (in place)
        k_col_stats<<<200, 128, 0, stream>>>(hnext, stats, stats + 128);
        k_bn_relu4<<<(ND4 + CPBLK - 1) / CPBLK, CPBLK, 0, stream>>>(
            (float4*)hnext, stats, stats + 128, g1, be1, L);

        // BN2 + ReLU (in place)
        k_col_stats<<<200, 128, 0, stream>>>(hnext, stats + 256, stats + 384);
        k_bn_relu4<<<(ND4 + CPBLK - 1) / CPBLK, CPBLK, 0, stream>>>(
            (float4*)hnext, stats + 256, stats + 384, g2, be2, L);
    }
}